// float_estimate_67448166416544
// MI455X (gfx1250) — compile-verified
//
#include <hip/hip_runtime.h>
#include <hip/hip_bf16.h>

namespace {

constexpr int kH = 256, kW = 256, kP = kH * kW;
constexpr int kNV = 6890;
constexpr int kNF = 13776;
constexpr int kRounds = 54;                 // 54 * 256 faces
constexpr int kNFPad = kRounds * 256;       // 13824 (>= 13776)
constexpr int kB = 2;
constexpr float kBig = 1000000000.0f;
constexpr float kEps = 1e-8f;

typedef __attribute__((ext_vector_type(2))) float v2f;
typedef __attribute__((ext_vector_type(8))) float v8f;

// Monotonic float -> uint key; ashr + (or^xor) fuses to v_bitop3 on gfx1250.
__device__ __forceinline__ unsigned depth_key(float f) {
  unsigned u = __float_as_uint(f);
  return u ^ ((unsigned)((int)u >> 31) | 0x80000000u);
}

// ---------------------------------------------------------------------------
// Kernel 1: per-face setup. Folds projection + 1/denom + z interpolation into
// affine edge coefficients so the rasterizer is pure [pix x 4] x [4 x face]:
//   w0(px,py) = a0*px + b0*py + c0        (cw0)
//   w1(px,py) = a1*px + b1*py + c1        (cw1)
//   w2(px,py) = 1 - w0 - w1               (cw2, explicit so w2 comes from WMMA)
//   depth     = ad*px + bd*py + cd        (cdp)
// Degenerate / padded faces can never pass the inside test.
// ---------------------------------------------------------------------------
__global__ void setup_faces_kernel(
    const float* __restrict__ src_cam, const float* __restrict__ src_verts,
    const float* __restrict__ tgt_cam, const float* __restrict__ tgt_verts,
    const int* __restrict__ faces,
    float* __restrict__ cw0, float* __restrict__ cw1, float* __restrict__ cw2,
    float* __restrict__ cdp, float* __restrict__ sf2v) {
  int idx = blockIdx.x * blockDim.x + threadIdx.x;
  if (idx >= kB * kNFPad) return;
  int b = idx / kNFPad;
  int f = idx - b * kNFPad;
  int o = idx * 4;
  if (f >= kNF) {  // padded slot: never wins
    cw0[o + 0] = 0.f; cw0[o + 1] = 0.f; cw0[o + 2] = -1.f; cw0[o + 3] = 0.f;
    cw1[o + 0] = 0.f; cw1[o + 1] = 0.f; cw1[o + 2] = -1.f; cw1[o + 3] = 0.f;
    cw2[o + 0] = 0.f; cw2[o + 1] = 0.f; cw2[o + 2] = -1.f; cw2[o + 3] = 0.f;
    cdp[o + 0] = 0.f; cdp[o + 1] = 0.f; cdp[o + 2] = kBig; cdp[o + 3] = 0.f;
    return;
  }
  int i0 = faces[3 * f + 0], i1 = faces[3 * f + 1], i2 = faces[3 * f + 2];
  const float* tc = tgt_cam + 3 * b;
  const float* tv = tgt_verts + (size_t)b * kNV * 3;
  float s = tc[0], txc = tc[1], tyc = tc[2];
  // _project: (s*(x+tx), -s*(y+ty), z)
  float x0 = s * (tv[3 * i0 + 0] + txc), y0 = -s * (tv[3 * i0 + 1] + tyc), z0 = tv[3 * i0 + 2];
  float x1 = s * (tv[3 * i1 + 0] + txc), y1 = -s * (tv[3 * i1 + 1] + tyc), z1 = tv[3 * i1 + 2];
  float x2 = s * (tv[3 * i2 + 0] + txc), y2 = -s * (tv[3 * i2 + 1] + tyc), z2 = tv[3 * i2 + 2];

  float denom = (y1 - y2) * (x0 - x2) + (x2 - x1) * (y0 - y2);
  bool valid = fabsf(denom) >= kEps;
  float inv = valid ? 1.0f / denom : 0.0f;
  float a0 = (y1 - y2) * inv, b0 = (x2 - x1) * inv;
  float c0 = -(a0 * x2 + b0 * y2);
  float a1 = (y2 - y0) * inv, b1 = (x0 - x2) * inv;
  float c1 = -(a1 * x2 + b1 * y2);
  float dz0 = z0 - z2, dz1 = z1 - z2;
  float ad = a0 * dz0 + a1 * dz1;
  float bd = b0 * dz0 + b1 * dz1;
  float cd = z2 + c0 * dz0 + c1 * dz1;
  if (!valid) {
    a0 = b0 = 0.f; c0 = -1.f;
    a1 = b1 = 0.f; c1 = -1.f;
    ad = bd = 0.f; cd = kBig;
  }
  float a2 = -(a0 + a1), b2 = -(b0 + b1), c2 = 1.0f - c0 - c1;
  cw0[o + 0] = a0; cw0[o + 1] = b0; cw0[o + 2] = c0; cw0[o + 3] = 0.f;
  cw1[o + 0] = a1; cw1[o + 1] = b1; cw1[o + 2] = c1; cw1[o + 3] = 0.f;
  cw2[o + 0] = a2; cw2[o + 1] = b2; cw2[o + 2] = c2; cw2[o + 3] = 0.f;
  cdp[o + 0] = ad; cdp[o + 1] = bd; cdp[o + 2] = cd; cdp[o + 3] = 0.f;

  // src_f2v: _project negates y, then reference multiplies y by -1 again,
  // so the net is simply s*(v + cam_offset) for both components.
  const float* sc = src_cam + 3 * b;
  const float* sv = src_verts + (size_t)b * kNV * 3;
  float ss = sc[0], sxc = sc[1], syc = sc[2];
  int so = (b * kNF + f) * 8;
  sf2v[so + 0] = ss * (sv[3 * i0 + 0] + sxc); sf2v[so + 1] = ss * (sv[3 * i0 + 1] + syc);
  sf2v[so + 2] = ss * (sv[3 * i1 + 0] + sxc); sf2v[so + 3] = ss * (sv[3 * i1 + 1] + syc);
  sf2v[so + 4] = ss * (sv[3 * i2 + 0] + sxc); sf2v[so + 5] = ss * (sv[3 * i2 + 1] + syc);
  sf2v[so + 6] = 0.f; sf2v[so + 7] = 0.f;
}

// ---------------------------------------------------------------------------
// Kernel 2: fused WMMA rasterizer + flow + border grid-sample.
// One wave owns a strip of 16 consecutive pixels. Per 16-face tile:
// four V_WMMA_F32_16X16X4_F32 (w0, w1, w2, depth). Z-buffer = per-lane packed
// u64 (depthkey<<32 | faceid) v_min_u64, folded across lane^16 at the end.
// Face-coefficient super-tiles (256 faces) are streamed global->LDS with
// double-buffered GLOBAL_LOAD_ASYNC_TO_LDS_B128 (ASYNCcnt) to hide L2 latency.
// ---------------------------------------------------------------------------
__global__ void __launch_bounds__(256)
raster_flow_sample_kernel(
    const float* __restrict__ src_img,
    const float* __restrict__ cw0, const float* __restrict__ cw1,
    const float* __restrict__ cw2, const float* __restrict__ cdp,
    const float* __restrict__ sf2v,
    float* __restrict__ out) {
  // [2 buffers][4 arrays][256 faces][4 coeffs]
  __shared__ float smem[2][4 * 256 * 4];

  const int tid = threadIdx.x;
  const int lane = tid & 31;
  const int n = lane & 15;     // pixel within strip / face within tile (A rows)
  const int hi = lane >> 4;    // K-half select per ISA A/B layouts
  const int waveInBlock = tid >> 5;

  const int b = blockIdx.x >> 9;                              // 512 blocks/batch
  const int strip = ((blockIdx.x & 511) << 3) + waveInBlock;  // 0..4095
  const int pb = strip << 4;
  const int row = pb >> 8;
  const int col0 = pb & 255;

  const float px = ((float)(col0 + n) + 0.5f) * (2.0f / 256.0f) - 1.0f;
  const float py = ((float)row + 0.5f) * (2.0f / 256.0f) - 1.0f;

  // B matrix (4 x 16): rows K = {px, py, 1, 0}.
  // Layout: VGPR v, lane l -> B[K = 2*v + (l>=16)][N = l%16].
  v2f Bm;
  Bm.x = hi ? py : px;    // K=0 -> px, K=1 -> py
  Bm.y = hi ? 0.f : 1.f;  // K=2 -> 1,  K=3 -> 0

  const float* gc0 = cw0 + (size_t)b * kNFPad * 4;
  const float* gc1 = cw1 + (size_t)b * kNFPad * 4;
  const float* gc2 = cw2 + (size_t)b * kNFPad * 4;
  const float* gc3 = cdp + (size_t)b * kNFPad * 4;

  const unsigned lds_base = (unsigned)(uintptr_t)&smem[0][0];

  // Issue one round's 4 B128 async global->LDS transfers (per thread).
  auto issue_round = [&](int round, int buf) {
    unsigned go = (unsigned)((round * 256 + tid) * 16);          // byte offset
    unsigned la = lds_base + (unsigned)buf * 16384u + (unsigned)tid * 16u;
    asm volatile("global_load_async_to_lds_b128 %0, %1, %2"
                 :: "v"(la + 0u), "v"(go), "s"(gc0) : "memory");
    asm volatile("global_load_async_to_lds_b128 %0, %1, %2"
                 :: "v"(la + 4096u), "v"(go), "s"(gc1) : "memory");
    asm volatile("global_load_async_to_lds_b128 %0, %1, %2"
                 :: "v"(la + 8192u), "v"(go), "s"(gc2) : "memory");
    asm volatile("global_load_async_to_lds_b128 %0, %1, %2"
                 :: "v"(la + 12288u), "v"(go), "s"(gc3) : "memory");
  };

  unsigned long long best = ~0ULL;

  issue_round(0, 0);
  for (int round = 0; round < kRounds; ++round) {
    const int buf = round & 1;
    if (round + 1 < kRounds) {
      issue_round(round + 1, buf ^ 1);
      // Done-signals return in order: <=4 outstanding => this round's 4 landed.
      asm volatile("s_wait_asynccnt 0x4" ::: "memory");
    } else {
      asm volatile("s_wait_asynccnt 0x0" ::: "memory");
    }
    __syncthreads();

    const float* sb = smem[buf];
    const int base = round * 256;
    #pragma unroll 4
    for (int tl = 0; tl < 16; ++tl) {
      // A fragment (16 faces x 4): lanes 0-15 hold K0,K1 of face n;
      // lanes 16-31 hold K2,K3 of face n.
      int fo = (tl * 16 + n) * 4 + hi * 2;
      v2f A0 = *(const v2f*)(sb + 0 * 1024 + fo);
      v2f A1 = *(const v2f*)(sb + 1 * 1024 + fo);
      v2f A2 = *(const v2f*)(sb + 2 * 1024 + fo);
      v2f A3 = *(const v2f*)(sb + 3 * 1024 + fo);
      v8f zero = {0.f, 0.f, 0.f, 0.f, 0.f, 0.f, 0.f, 0.f};
      v8f W0 = __builtin_amdgcn_wmma_f32_16x16x4_f32(
          false, A0, false, Bm, (short)0, zero, false, false);
      v8f W1 = __builtin_amdgcn_wmma_f32_16x16x4_f32(
          false, A1, false, Bm, (short)0, zero, false, false);
      v8f W2 = __builtin_amdgcn_wmma_f32_16x16x4_f32(
          false, A2, false, Bm, (short)0, zero, false, false);
      v8f DD = __builtin_amdgcn_wmma_f32_16x16x4_f32(
          false, A3, false, Bm, (short)0, zero, false, false);

      // D layout: lane l, VGPR r -> D[M = r + 8*(l>=16)][N = l%16]
      unsigned fbase = (unsigned)(base + tl * 16 + hi * 8);
      #pragma unroll
      for (int r = 0; r < 8; ++r) {
        float m = fminf(fminf(W0[r], W1[r]), W2[r]);  // v_min3_num_f32
        // Select in float domain (both operands are plain values -> single
        // branchless v_cndmask with the 1e9 literal), THEN transform.
        float d = (m >= 0.f) ? DD[r] : kBig;
        unsigned dkey = depth_key(d);                 // v_ashrrev + v_bitop3
        unsigned long long key =
            ((unsigned long long)dkey << 32) | (fbase + (unsigned)r);
        best = (key < best) ? key : best;             // v_min_u64
      }
    }
    __syncthreads();  // all waves done reading smem[buf] before it is refilled
  }

  // Fold faces handled by lane^16 (same pixel N = lane%16).
  unsigned blo = (unsigned)best;
  unsigned bhi = (unsigned)(best >> 32);
  unsigned olo = __shfl_xor(blo, 16, 32);
  unsigned ohi = __shfl_xor(bhi, 16, 32);
  unsigned long long other = ((unsigned long long)ohi << 32) | olo;
  best = (other < best) ? other : best;

  if (hi == 0) {
    const unsigned bigkey = depth_key(kBig);
    unsigned dk = (unsigned)(best >> 32);
    int fid = (int)(best & 0xffffffffu);
    float fx, fy;
    if (dk < bigkey) {
      const float* c0p = cw0 + ((size_t)b * kNFPad + fid) * 4;
      const float* c1p = cw1 + ((size_t)b * kNFPad + fid) * 4;
      float w0 = c0p[0] * px + c0p[1] * py + c0p[2];
      float w1 = c1p[0] * px + c1p[1] * py + c1p[2];
      float w2 = 1.0f - w0 - w1;
      const float* sp = sf2v + ((size_t)b * kNF + fid) * 8;
      fx = w0 * sp[0] + w1 * sp[2] + w2 * sp[4];
      fy = w0 * sp[1] + w1 * sp[3] + w2 * sp[5];
    } else {
      fx = -2.0f;
      fy = -2.0f;
    }
    // grid_sample, border padding
    float ix = fminf(fmaxf(((fx + 1.0f) * 256.0f - 1.0f) * 0.5f, 0.0f), 255.0f);
    float iy = fminf(fmaxf(((fy + 1.0f) * 256.0f - 1.0f) * 0.5f, 0.0f), 255.0f);
    float xf = floorf(ix), yf = floorf(iy);
    int x0i = (int)xf, y0i = (int)yf;
    float wx = ix - xf, wy = iy - yf;
    int x1i = min(x0i + 1, 255);
    int y1i = min(y0i + 1, 255);
    float wa = (1.0f - wx) * (1.0f - wy), wb = wx * (1.0f - wy);
    float wc = (1.0f - wx) * wy, wd = wx * wy;
    const float* img = src_img + (size_t)b * 3 * kP;
    int p = pb + n;
    #pragma unroll
    for (int ch = 0; ch < 3; ++ch) {
      const float* ic = img + ch * kP;
      float v = ic[y0i * 256 + x0i] * wa + ic[y0i * 256 + x1i] * wb +
                ic[y1i * 256 + x0i] * wc + ic[y1i * 256 + x1i] * wd;
      out[(size_t)b * 3 * kP + (size_t)ch * kP + p] = v;
    }
  }
}

}  // namespace

extern "C" void kernel_launch(void* const* d_in, const int* in_sizes, int n_in,
                              void* d_out, int out_size, void* d_ws, size_t ws_size,
                              hipStream_t stream) {
  const float* src_img   = (const float*)d_in[0];
  const float* src_cam   = (const float*)d_in[1];
  const float* src_verts = (const float*)d_in[2];
  const float* tgt_cam   = (const float*)d_in[3];
  const float* tgt_verts = (const float*)d_in[4];
  const int*   faces     = (const int*)d_in[5];

  // Workspace layout (floats): cw0 | cw1 | cw2 | cd | sf2v  (~2.7 MB total)
  float* ws = (float*)d_ws;
  float* cw0  = ws;
  float* cw1  = cw0 + (size_t)kB * kNFPad * 4;
  float* cw2  = cw1 + (size_t)kB * kNFPad * 4;
  float* cdp  = cw2 + (size_t)kB * kNFPad * 4;
  float* sf2v = cdp + (size_t)kB * kNFPad * 4;

  int total = kB * kNFPad;
  setup_faces_kernel<<<(total + 255) / 256, 256, 0, stream>>>(
      src_cam, src_verts, tgt_cam, tgt_verts, faces, cw0, cw1, cw2, cdp, sf2v);

  // 2 batches * 4096 strips = 8192 waves = 1024 blocks of 8 waves
  raster_flow_sample_kernel<<<kB * 512, 256, 0, stream>>>(
      src_img, cw0, cw1, cw2, cdp, sf2v, (float*)d_out);
}